// GraphSelfAttentionNetwork_80633716015118
// MI455X (gfx1250) — compile-verified
//
#include <hip/hip_runtime.h>

typedef __attribute__((ext_vector_type(2))) float v2f;
typedef __attribute__((ext_vector_type(8))) float v8f;

#define N_NODES   50000
#define N_EDGES   800000
#define D_MODEL   128
#define N_HEADS   8
#define HEAD_DIM  16
#define QKV_COLS  384
#define LN_EPS    1e-5f
#define QSCALE    0.08838834764831845f   // 128^-0.5

// ---------------------------------------------------------------------------
// Init workspace: agg = 0, m = -inf, denom = 0 (must run every call)
// ---------------------------------------------------------------------------
__global__ __launch_bounds__(256) void gsan_init_kernel(float* __restrict__ m,
                                                        float* __restrict__ denom,
                                                        float* __restrict__ agg) {
  int i = blockIdx.x * 256 + threadIdx.x;
  if (i < N_NODES * D_MODEL) agg[i] = 0.0f;
  if (i < N_NODES * N_HEADS) { m[i] = -__builtin_inff(); denom[i] = 0.0f; }
}

// ---------------------------------------------------------------------------
// Kernel A: fused LayerNorm + QKV GEMM using V_WMMA_F32_16X16X4_F32.
// One block = 16 nodes. 8 waves, each wave computes 3 of the 24 N-tiles.
// ---------------------------------------------------------------------------
__global__ __launch_bounds__(256) void gsan_ln_qkv_kernel(
    const float* __restrict__ node_feature,   // [N_NODES][128]
    const float* __restrict__ gamma,          // [128]
    const float* __restrict__ beta,           // [128]
    const float* __restrict__ W_qkv,          // [384][128]
    const float* __restrict__ b_qkv,          // [384]
    float* __restrict__ qkv)                  // [N_NODES][384]  (q pre-scaled)
{
  __shared__ float xs[16][132];               // pad 132: avoid 64-bank conflicts
  __shared__ float smu[16], srs[16];
  const int tid = threadIdx.x;
  const int nodeBase = blockIdx.x * 16;

  // cooperative load of the 16x128 tile
  for (int i = tid; i < 16 * 128; i += 256) {
    int r = i >> 7, c = i & 127;
    xs[r][c] = node_feature[(nodeBase + r) * D_MODEL + c];
  }
  __syncthreads();

  // per-row mean / population-variance (matches jnp.var, ddof=0)
  if (tid < 16) {
    float s = 0.0f, sq = 0.0f;
    for (int c = 0; c < 128; ++c) { float v = xs[tid][c]; s += v; sq += v * v; }
    float mu  = s * (1.0f / 128.0f);
    float var = sq * (1.0f / 128.0f) - mu * mu;
    smu[tid] = mu;
    srs[tid] = rsqrtf(var + LN_EPS);
  }
  __syncthreads();

  for (int i = tid; i < 16 * 128; i += 256) {
    int r = i >> 7, c = i & 127;
    xs[r][c] = (xs[r][c] - smu[r]) * srs[r] * gamma[c] + beta[c];
  }
  __syncthreads();

  const int lane = tid & 31;
  const int wave = tid >> 5;
  const int row  = lane & 15;   // A-matrix M row / B-matrix N column (lane-local)
  const int half = lane >> 4;   // K-half select per ISA VGPR layout

  for (int t = 0; t < 3; ++t) {
    const int nt  = wave + t * 8;               // N-tile 0..23
    const int col = nt * 16 + row;              // output column for this lane
    const float* wrow = W_qkv + col * D_MODEL;  // B[k][n] = W_qkv[n][k]
    v8f acc = {};
    #pragma unroll 8
    for (int k0 = 0; k0 < D_MODEL; k0 += 4) {
      const int k = k0 + 2 * half;
      v2f a, b;
      a.x = xs[row][k];  a.y = xs[row][k + 1];  // A 16x4: V0=K+0 / V1=K+1 (half: +2)
      b.x = wrow[k];     b.y = wrow[k + 1];     // B 4x16: same K striping, N on lanes
      acc = __builtin_amdgcn_wmma_f32_16x16x4_f32(
          false, a, false, b, (short)0, acc, false, false);
    }
    const float bias  = b_qkv[col];
    const float scale = (nt < 8) ? QSCALE : 1.0f;  // q block scaled AFTER bias
    #pragma unroll
    for (int j = 0; j < 8; ++j) {               // C layout: M = j + 8*half, N = row
      const int mrow = j + 8 * half;
      qkv[(nodeBase + mrow) * QKV_COLS + col] = (acc[j] + bias) * scale;
    }
  }
}

// ---------------------------------------------------------------------------
// Edge phase
// ---------------------------------------------------------------------------
__device__ inline void atomicMaxF(float* addr, float val) {
  if (val >= 0.0f) atomicMax((int*)addr, __float_as_int(val));
  else             atomicMin((unsigned int*)addr, __float_as_uint(val));
}

__global__ __launch_bounds__(256) void gsan_edge_attn_kernel(
    const float* __restrict__ qkv,
    const float* __restrict__ dist_attn,
    const float* __restrict__ path_attn,
    const int*   __restrict__ src,
    const int*   __restrict__ dst,
    float* __restrict__ a_ws,
    float* __restrict__ m)
{
  int i = blockIdx.x * 256 + threadIdx.x;
  if (i >= N_EDGES * N_HEADS) return;
  const int e = i >> 3, h = i & 7;
  const int s = src[e], d = dst[e];
  const float4* qp = (const float4*)(qkv + (size_t)s * QKV_COLS +       h * HEAD_DIM);
  const float4* kp = (const float4*)(qkv + (size_t)d * QKV_COLS + 128 + h * HEAD_DIM);
  float acc = 0.0f;
  #pragma unroll
  for (int j = 0; j < 4; ++j) {
    float4 q4 = qp[j], k4 = kp[j];
    acc += q4.x * k4.x + q4.y * k4.y + q4.z * k4.z + q4.w * k4.w;
  }
  const float a = acc + dist_attn[i] + path_attn[i];
  a_ws[i] = a;
  atomicMaxF(&m[d * N_HEADS + h], a);
}

__global__ __launch_bounds__(256) void gsan_edge_exp_kernel(
    const int*   __restrict__ dst,
    const float* __restrict__ m,
    float* __restrict__ a_ws,
    float* __restrict__ denom)
{
  int i = blockIdx.x * 256 + threadIdx.x;
  if (i >= N_EDGES * N_HEADS) return;
  const int e = i >> 3, h = i & 7;
  const int d = dst[e];
  const float ex = expf(a_ws[i] - m[d * N_HEADS + h]);
  a_ws[i] = ex;
  atomicAdd(&denom[d * N_HEADS + h], ex);
}

__global__ __launch_bounds__(256) void gsan_edge_agg_kernel(
    const int*   __restrict__ src,
    const int*   __restrict__ dst,
    const float* __restrict__ qkv,
    const float* __restrict__ a_ws,
    const float* __restrict__ denom,
    float* __restrict__ agg)
{
  int i = blockIdx.x * 256 + threadIdx.x;
  if (i >= N_EDGES * N_HEADS) return;
  const int e = i >> 3, h = i & 7;
  const int s = src[e], d = dst[e];
  const float sa = a_ws[i] / denom[d * N_HEADS + h];
  const float* vp = qkv + (size_t)s * QKV_COLS + 256 + h * HEAD_DIM;
  float* ap = agg + (size_t)d * D_MODEL + h * HEAD_DIM;
  #pragma unroll
  for (int j = 0; j < HEAD_DIM; ++j) atomicAdd(&ap[j], vp[j] * sa);
}

// ---------------------------------------------------------------------------
// Kernel C: out = node_feature + agg @ W_out^T + b_out  (WMMA f32 16x16x4)
// One block = 16 nodes; 8 waves x one 16-wide N-tile each (128 cols).
// ---------------------------------------------------------------------------
__global__ __launch_bounds__(256) void gsan_out_gemm_kernel(
    const float* __restrict__ agg,
    const float* __restrict__ W_out,   // [128][128]
    const float* __restrict__ b_out,   // [128]
    const float* __restrict__ node_feature,
    float* __restrict__ out)
{
  __shared__ float xs[16][132];
  const int tid = threadIdx.x;
  const int nodeBase = blockIdx.x * 16;

  for (int i = tid; i < 16 * 128; i += 256) {
    int r = i >> 7, c = i & 127;
    xs[r][c] = agg[(nodeBase + r) * D_MODEL + c];
  }
  __syncthreads();

  const int lane = tid & 31;
  const int wave = tid >> 5;
  const int row  = lane & 15;
  const int half = lane >> 4;
  const int col  = wave * 16 + row;
  const float* wrow = W_out + col * D_MODEL;

  v8f acc = {};
  #pragma unroll 8
  for (int k0 = 0; k0 < D_MODEL; k0 += 4) {
    const int k = k0 + 2 * half;
    v2f a, b;
    a.x = xs[row][k];  a.y = xs[row][k + 1];
    b.x = wrow[k];     b.y = wrow[k + 1];
    acc = __builtin_amdgcn_wmma_f32_16x16x4_f32(
        false, a, false, b, (short)0, acc, false, false);
  }
  const float bias = b_out[col];
  #pragma unroll
  for (int j = 0; j < 8; ++j) {
    const int mrow = j + 8 * half;
    const int idx  = (nodeBase + mrow) * D_MODEL + col;
    out[idx] = node_feature[idx] + acc[j] + bias;
  }
}

// ---------------------------------------------------------------------------
// Launch
// ---------------------------------------------------------------------------
extern "C" void kernel_launch(void* const* d_in, const int* in_sizes, int n_in,
                              void* d_out, int out_size, void* d_ws, size_t ws_size,
                              hipStream_t stream) {
  const float* node_feature = (const float*)d_in[0];
  const float* dist_attn    = (const float*)d_in[1];
  const float* path_attn    = (const float*)d_in[2];
  const float* gamma        = (const float*)d_in[3];
  const float* beta         = (const float*)d_in[4];
  const float* W_qkv        = (const float*)d_in[5];
  const float* b_qkv        = (const float*)d_in[6];
  const float* W_out        = (const float*)d_in[7];
  const float* b_out        = (const float*)d_in[8];
  const int*   src          = (const int*)d_in[9];
  const int*   dst          = (const int*)d_in[10];
  float*       out          = (float*)d_out;

  // Workspace partition (floats):
  //   qkv   : N_NODES*384        = 19,200,000
  //   a_ws  : N_EDGES*8          =  6,400,000
  //   m     : N_NODES*8          =    400,000
  //   denom : N_NODES*8          =    400,000
  //   agg   : N_NODES*128        =  6,400,000   -> total 131.2 MB
  float* ws    = (float*)d_ws;
  float* qkv   = ws;
  float* a_ws  = qkv   + (size_t)N_NODES * QKV_COLS;
  float* m     = a_ws  + (size_t)N_EDGES * N_HEADS;
  float* denom = m     + (size_t)N_NODES * N_HEADS;
  float* agg   = denom + (size_t)N_NODES * N_HEADS;

  const int edgeBlocks = (N_EDGES * N_HEADS + 255) / 256;   // 25000
  const int nodeBlocks = N_NODES / 16;                      // 3125 (exact)
  const int initBlocks = (N_NODES * D_MODEL + 255) / 256;   // 25000

  gsan_init_kernel<<<initBlocks, 256, 0, stream>>>(m, denom, agg);
  gsan_ln_qkv_kernel<<<nodeBlocks, 256, 0, stream>>>(
      node_feature, gamma, beta, W_qkv, b_qkv, qkv);
  gsan_edge_attn_kernel<<<edgeBlocks, 256, 0, stream>>>(
      qkv, dist_attn, path_attn, src, dst, a_ws, m);
  gsan_edge_exp_kernel<<<edgeBlocks, 256, 0, stream>>>(dst, m, a_ws, denom);
  gsan_edge_agg_kernel<<<edgeBlocks, 256, 0, stream>>>(
      src, dst, qkv, a_ws, denom, agg);
  gsan_out_gemm_kernel<<<nodeBlocks, 256, 0, stream>>>(
      agg, W_out, b_out, node_feature, out);
}